// GatedDeltaNet_16862041604483
// MI455X (gfx1250) — compile-verified
//
#include <hip/hip_runtime.h>
#include <math.h>

typedef __attribute__((ext_vector_type(2))) float v2f;
typedef __attribute__((ext_vector_type(8))) float v8f;
typedef __attribute__((ext_vector_type(4))) int   v4i;

#define HQ 14
#define HKV 2
#define GG 7
#define DD 64
#define HID 896
#define NB 2
#define NN 2048
#define TOK (NB * NN)   // 4096

#define AS1 __attribute__((address_space(1)))
#define AS3 __attribute__((address_space(3)))

// ---- CDNA5 async global->LDS path (guarded: falls back to sync staging) ----
#if defined(__gfx1250__) && \
    __has_builtin(__builtin_amdgcn_global_load_async_to_lds_b128) && \
    __has_builtin(__builtin_amdgcn_global_load_async_to_lds_b32)
#define USE_ASYNC 1
#else
#define USE_ASYNC 0
#endif

#if USE_ASYNC
__device__ __forceinline__ void async_cp16(const float* g, float* l) {
    __builtin_amdgcn_global_load_async_to_lds_b128(
        (AS1 v4i*)(uintptr_t)g,
        (AS3 v4i*)(unsigned)(uintptr_t)l, 0, 0);
}
__device__ __forceinline__ void async_cp4(const float* g, float* l) {
    __builtin_amdgcn_global_load_async_to_lds_b32(
        (AS1 int*)(uintptr_t)g,
        (AS3 int*)(unsigned)(uintptr_t)l, 0, 0);
}
#if __has_builtin(__builtin_amdgcn_s_wait_asynccnt)
#define WAIT_ASYNC(n) __builtin_amdgcn_s_wait_asynccnt(n)
#else
#define WAIT_ASYNC(n) asm volatile("s_wait_asynccnt " #n ::: "memory")
#endif
#else
#define WAIT_ASYNC(n)
#endif

__device__ __forceinline__ float sigmoidf_(float x) { return 1.0f / (1.0f + expf(-x)); }

// ---------------------------------------------------------------------------
// Tiled fp32 GEMM on the WMMA pipe: C[M,Ncols] = X[M,K] @ W[K,Ncols] (+bias, act)
// Block tile 64(M) x 128(N), K panel 32, double-buffered async LDS staging.
// 8 waves, each wave owns 16x64 (4 x v_wmma_f32_16x16x4_f32 per k-step).
// act: 0 = none, 1 = sigmoid
// ---------------------------------------------------------------------------
__global__ __launch_bounds__(256) void gemm_wmma_f32(
    const float* __restrict__ X, const float* __restrict__ W,
    const float* __restrict__ bias, float* __restrict__ C,
    int K, int Ncols, int act)
{
    // strides 36/136: conflict-free A reads, disjoint half-wave B banks,
    // rows remain 16B multiples so async B128 LDS writes stay aligned.
    __shared__ float __attribute__((aligned(16))) As[2][64][36];
    __shared__ float __attribute__((aligned(16))) Bs[2][32][136];

    const int t    = threadIdx.x;
    const int wave = t >> 5;
    const int lane = t & 31;
    const int wm   = wave >> 1;      // 0..3  -> 16-row subtile
    const int wn   = wave & 1;       // 0..1  -> 64-col half
    const int lm   = lane & 15;
    const int lh   = lane >> 4;      // half-wave select

    const int rowBase = blockIdx.x * 64;
    const int colBase = blockIdx.y * 128;

    v8f acc0 = {}, acc1 = {}, acc2 = {}, acc3 = {};

    const int alr = t >> 2;          // A load row   0..63
    const int alc = (t & 3) * 8;     // A load col   0,8,16,24
    const int blr = t >> 3;          // B load row   0..31
    const int blc = (t & 7) * 16;    // B load col   0..112

    const float* aSrc = X + (size_t)(rowBase + alr) * K + alc;
    const float* bSrc = W + (size_t)blr * Ncols + colBase + blc;

    auto prefetch = [&](int kb, int buf) {
#if USE_ASYNC
        async_cp16(aSrc + kb,     &As[buf][alr][alc]);
        async_cp16(aSrc + kb + 4, &As[buf][alr][alc + 4]);
        const float* bp = bSrc + (size_t)kb * Ncols;
        async_cp16(bp + 0,  &Bs[buf][blr][blc + 0]);
        async_cp16(bp + 4,  &Bs[buf][blr][blc + 4]);
        async_cp16(bp + 8,  &Bs[buf][blr][blc + 8]);
        async_cp16(bp + 12, &Bs[buf][blr][blc + 12]);
#else
        const float4* sa = reinterpret_cast<const float4*>(aSrc + kb);
        float4 a0 = sa[0], a1 = sa[1];
        As[buf][alr][alc + 0] = a0.x; As[buf][alr][alc + 1] = a0.y;
        As[buf][alr][alc + 2] = a0.z; As[buf][alr][alc + 3] = a0.w;
        As[buf][alr][alc + 4] = a1.x; As[buf][alr][alc + 5] = a1.y;
        As[buf][alr][alc + 6] = a1.z; As[buf][alr][alc + 7] = a1.w;
        const float4* sb = reinterpret_cast<const float4*>(bSrc + (size_t)kb * Ncols);
        #pragma unroll
        for (int q = 0; q < 4; ++q) {
            float4 b4 = sb[q];
            Bs[buf][blr][blc + q * 4 + 0] = b4.x;
            Bs[buf][blr][blc + q * 4 + 1] = b4.y;
            Bs[buf][blr][blc + q * 4 + 2] = b4.z;
            Bs[buf][blr][blc + q * 4 + 3] = b4.w;
        }
#endif
    };

    const int nPan = K >> 5;
    prefetch(0, 0);
    for (int p = 0; p < nPan; ++p) {
        const int buf = p & 1;
        if (p + 1 < nPan) {
            prefetch((p + 1) << 5, buf ^ 1);   // stream next panel while computing
            WAIT_ASYNC(6);                     // our 6 panel-p ops complete
        } else {
            WAIT_ASYNC(0);
        }
        __syncthreads();

        #pragma unroll
        for (int kk = 0; kk < 32; kk += 4) {
            v2f a;
            a.x = As[buf][wm * 16 + lm][kk + lh * 2 + 0];
            a.y = As[buf][wm * 16 + lm][kk + lh * 2 + 1];

            v2f bfr;
            int c0 = wn * 64 + lm;
            bfr.x = Bs[buf][kk + lh * 2 + 0][c0];
            bfr.y = Bs[buf][kk + lh * 2 + 1][c0];
            acc0 = __builtin_amdgcn_wmma_f32_16x16x4_f32(false, a, false, bfr, (short)0, acc0, false, false);

            bfr.x = Bs[buf][kk + lh * 2 + 0][c0 + 16];
            bfr.y = Bs[buf][kk + lh * 2 + 1][c0 + 16];
            acc1 = __builtin_amdgcn_wmma_f32_16x16x4_f32(false, a, false, bfr, (short)0, acc1, false, false);

            bfr.x = Bs[buf][kk + lh * 2 + 0][c0 + 32];
            bfr.y = Bs[buf][kk + lh * 2 + 1][c0 + 32];
            acc2 = __builtin_amdgcn_wmma_f32_16x16x4_f32(false, a, false, bfr, (short)0, acc2, false, false);

            bfr.x = Bs[buf][kk + lh * 2 + 0][c0 + 48];
            bfr.y = Bs[buf][kk + lh * 2 + 1][c0 + 48];
            acc3 = __builtin_amdgcn_wmma_f32_16x16x4_f32(false, a, false, bfr, (short)0, acc3, false, false);
        }
        __syncthreads();
    }

    // epilogue: C/D layout: vgpr r -> M = r + lh*8, lane lm -> N
    v8f accs[4] = {acc0, acc1, acc2, acc3};
    #pragma unroll
    for (int nt = 0; nt < 4; ++nt) {
        int col = colBase + wn * 64 + nt * 16 + lm;
        float bv = bias ? bias[col] : 0.0f;
        #pragma unroll
        for (int r = 0; r < 8; ++r) {
            int row = rowBase + wm * 16 + lh * 8 + r;
            float v = accs[nt][r] + bv;
            if (act == 1) v = sigmoidf_(v);
            C[(size_t)row * Ncols + col] = v;
        }
    }
}

// ---------------------------------------------------------------------------
// Beta projection: out[tok,h] = sigmoid(X[tok,:] @ Wbeta[:,h] + bbeta[h]) (Ncols=2)
// ---------------------------------------------------------------------------
__global__ void beta_proj(const float* __restrict__ X, const float* __restrict__ Wb,
                          const float* __restrict__ bb, float* __restrict__ out)
{
    int idx = blockIdx.x * blockDim.x + threadIdx.x;
    if (idx >= TOK * HKV) return;
    int tok = idx >> 1, h = idx & 1;
    float acc = bb[h];
    const float* xr = X + (size_t)tok * HID;
    for (int k = 0; k < HID; ++k) acc += xr[k] * Wb[k * HKV + h];
    out[idx] = sigmoidf_(acc);
}

// ---------------------------------------------------------------------------
// In-place RoPE + elu(x)+1 feature map on [TOK, nheads*64].
// One thread handles a (d, d+32) rotation pair of one head of one token.
// ---------------------------------------------------------------------------
__global__ void rope_elu(float* __restrict__ Xh, const int* __restrict__ pos, int nheads)
{
    int idx = blockIdx.x * blockDim.x + threadIdx.x;
    int total = TOK * nheads * 32;
    if (idx >= total) return;
    int p   = idx & 31;
    int hd  = (idx >> 5) % nheads;
    int tok = idx / (32 * nheads);

    float posv = (float)pos[tok];
    float freq = posv * expf(-((float)(2 * p) / 64.0f) * logf(1000000.0f));
    float cs = cosf(freq), sn = sinf(freq);

    size_t base = (size_t)tok * nheads * DD + (size_t)hd * DD;
    float x1 = Xh[base + p];
    float x2 = Xh[base + p + 32];
    float o1 = x1 * cs - x2 * sn;   // rotate_half: first half gets -x[d+32]
    float o2 = x2 * cs + x1 * sn;
    o1 = (o1 > 0.0f) ? o1 + 1.0f : expf(o1);
    o2 = (o2 > 0.0f) ? o2 + 1.0f : expf(o2);
    Xh[base + p]      = o1;
    Xh[base + p + 32] = o2;
}

// ---------------------------------------------------------------------------
// Sequential gated-delta scan. 4 blocks = (B, Hkv) chains, 256 threads each.
// Thread t owns  S [i=t>>2][ (t&3)*16 .. +16 )   (row copy, contract over j)
//          and  ST[j=t>>2][ (t&3)*16 .. +16 )   (transposed copy, contract over i)
// so both contractions are 16 local FMAs + 2 shfl_xor steps over 4 lanes.
// Per-step operands (k,v,a,q,beta) are double-buffered in LDS and streamed
// one step ahead with async global->LDS loads to hide latency on the
// serialized critical path.
// ---------------------------------------------------------------------------
__global__ __launch_bounds__(256) void scan_kernel(
    const float* __restrict__ Qf, const float* __restrict__ Kf,
    const float* __restrict__ Vp, const float* __restrict__ Ap,
    const float* __restrict__ Beta, const float* __restrict__ Gp,
    float* __restrict__ Oscan, float* __restrict__ Sfinal)
{
    const int b = blockIdx.x >> 1;
    const int h = blockIdx.x & 1;

    __shared__ float __attribute__((aligned(16))) lk[2][DD];
    __shared__ float __attribute__((aligned(16))) lv[2][DD];
    __shared__ float __attribute__((aligned(16))) la[2][DD];
    __shared__ float __attribute__((aligned(16))) lq[2][GG * DD];
    __shared__ float __attribute__((aligned(16))) lbeta[2];
    __shared__ float ldelta[DD];

    const int t  = threadIdx.x;
    const int r  = t >> 2;          // owned row of S / ST
    const int c4 = t & 3;
    const int j0 = c4 * 16;

    float S[16], ST[16];
    #pragma unroll
    for (int m = 0; m < 16; ++m) { S[m] = 0.0f; ST[m] = 0.0f; }

#if USE_ASYNC
    auto stage_async = [&](int n2, int buf) {
        const int tok2 = b * NN + n2;
        const size_t kvBase = (size_t)tok2 * (HKV * DD) + h * DD;
        if (t < 16)       async_cp16(Kf + kvBase + t * 4,        &lk[buf][t * 4]);
        else if (t < 32)  async_cp16(Vp + kvBase + (t - 16) * 4, &lv[buf][(t - 16) * 4]);
        else if (t < 48)  async_cp16(Ap + kvBase + (t - 32) * 4, &la[buf][(t - 32) * 4]);
        else if (t < 160) async_cp16(Qf + (size_t)tok2 * (HQ * DD) + h * (GG * DD) + (t - 48) * 4,
                                     &lq[buf][(t - 48) * 4]);
        else if (t == 160) async_cp4(Beta + tok2 * HKV + h, &lbeta[buf]);
    };
#endif

    for (int n = 0; n < NN; ++n) {
        const int tok = b * NN + n;
        const int buf = n & 1;
#if USE_ASYNC
        if (n == 0) { stage_async(0, 0); WAIT_ASYNC(0); }
        if (n + 1 < NN) { stage_async(n + 1, buf ^ 1); WAIT_ASYNC(1); }
        else            { WAIT_ASYNC(0); }
        __syncthreads();
#else
        if (t < DD) {
            size_t base = (size_t)tok * (HKV * DD) + h * DD;
            lk[buf][t] = Kf[base + t];
            lv[buf][t] = Vp[base + t];
            la[buf][t] = Ap[base + t];
        }
        if (t == 255) lbeta[buf] = Beta[tok * HKV + h];
        for (int i = t; i < GG * DD; i += 256)
            lq[buf][i] = Qf[(size_t)tok * (HQ * DD) + h * (GG * DD) + i];
        __syncthreads();
#endif

        const float bb = lbeta[buf];

        // retrieved[r] = sum_j S[r,j] * k[j]
        float part = 0.0f;
        #pragma unroll
        for (int m = 0; m < 16; ++m) part += S[m] * lk[buf][j0 + m];
        part += __shfl_xor(part, 1, 32);
        part += __shfl_xor(part, 2, 32);
        const float delta_r = lv[buf][r] - bb * part;   // delta[r]
        if (c4 == 0) ldelta[r] = delta_r;
        __syncthreads();

        // S[i,j]  = a[i]*S[i,j]  + b*k[i]*delta[j]
        const float ar = la[buf][r];
        const float bk = bb * lk[buf][r];
        #pragma unroll
        for (int m = 0; m < 16; ++m) S[m] = ar * S[m] + bk * ldelta[j0 + m];
        // ST[j,i] = a[i]*ST[j,i] + b*k[i]*delta[j]   (j = r fixed, i = j0+m)
        #pragma unroll
        for (int m = 0; m < 16; ++m) {
            int i = j0 + m;
            ST[m] = la[buf][i] * ST[m] + bb * lk[buf][i] * delta_r;
        }

        // o[g, j=r] = sum_i ST[r,i] * q[g,i], gated store
        #pragma unroll
        for (int g = 0; g < GG; ++g) {
            float po = 0.0f;
            #pragma unroll
            for (int m = 0; m < 16; ++m) po += ST[m] * lq[buf][g * DD + j0 + m];
            po += __shfl_xor(po, 1, 32);
            po += __shfl_xor(po, 2, 32);
            if (c4 == 0) {
                int hq = h * GG + g;
                size_t oi = (size_t)tok * (HQ * DD) + hq * DD + r;
                Oscan[oi] = po * Gp[oi];
            }
        }
        __syncthreads();   // protect buffers before next iteration's staging
    }

    // final state, layout [B, Hkv, D, D] row-major (i, j)
    #pragma unroll
    for (int m = 0; m < 16; ++m)
        Sfinal[(((size_t)(b * HKV + h) * DD) + r) * DD + j0 + m] = S[m];
}

// ---------------------------------------------------------------------------
extern "C" void kernel_launch(void* const* d_in, const int* in_sizes, int n_in,
                              void* d_out, int out_size, void* d_ws, size_t ws_size,
                              hipStream_t stream) {
    (void)in_sizes; (void)n_in; (void)out_size; (void)ws_size;
    const float* X      = (const float*)d_in[0];
    const int*   pos    = (const int*)  d_in[1];
    const float* Wq     = (const float*)d_in[2];
    const float* Wk     = (const float*)d_in[3];
    const float* Wv     = (const float*)d_in[4];
    const float* Wo     = (const float*)d_in[5];
    const float* Walpha = (const float*)d_in[6];
    const float* balpha = (const float*)d_in[7];
    const float* Wbeta  = (const float*)d_in[8];
    const float* bbeta  = (const float*)d_in[9];
    const float* Wgate  = (const float*)d_in[10];

    float* out   = (float*)d_out;
    float* outS  = out + (size_t)TOK * HID;          // S_final after main output

    float* ws = (float*)d_ws;
    float* Qp   = ws;                                 // TOK*896
    float* Gp   = Qp + (size_t)TOK * HID;             // TOK*896
    float* Kp   = Gp + (size_t)TOK * HID;             // TOK*128
    float* Vp   = Kp + (size_t)TOK * HKV * DD;        // TOK*128
    float* Ap   = Vp + (size_t)TOK * HKV * DD;        // TOK*128
    float* Bt   = Ap + (size_t)TOK * HKV * DD;        // TOK*2
    float* Osc  = Bt + (size_t)TOK * HKV;             // TOK*896

    // projections (WMMA fp32 GEMMs)
    gemm_wmma_f32<<<dim3(TOK / 64, HID / 128), 256, 0, stream>>>(X, Wq,     nullptr, Qp, HID, HID,      0);
    gemm_wmma_f32<<<dim3(TOK / 64, 1),         256, 0, stream>>>(X, Wk,     nullptr, Kp, HID, HKV * DD, 0);
    gemm_wmma_f32<<<dim3(TOK / 64, 1),         256, 0, stream>>>(X, Wv,     nullptr, Vp, HID, HKV * DD, 0);
    gemm_wmma_f32<<<dim3(TOK / 64, 1),         256, 0, stream>>>(X, Walpha, balpha,  Ap, HID, HKV * DD, 1);
    gemm_wmma_f32<<<dim3(TOK / 64, HID / 128), 256, 0, stream>>>(X, Wgate,  nullptr, Gp, HID, HID,      1);
    beta_proj<<<(TOK * HKV + 255) / 256, 256, 0, stream>>>(X, Wbeta, bbeta, Bt);

    // RoPE + elu+1 feature maps (in place)
    rope_elu<<<(TOK * HQ  * 32 + 255) / 256, 256, 0, stream>>>(Qp, pos, HQ);
    rope_elu<<<(TOK * HKV * 32 + 255) / 256, 256, 0, stream>>>(Kp, pos, HKV);

    // sequential gated-delta scan (4 independent chains), gate fused at store
    scan_kernel<<<NB * HKV, 256, 0, stream>>>(Qp, Kp, Vp, Ap, Bt, Gp, Osc, outS);

    // output projection
    gemm_wmma_f32<<<dim3(TOK / 64, HID / 128), 256, 0, stream>>>(Osc, Wo, nullptr, out, HID, HID, 0);
}